// Postprocess_29540785062349
// MI455X (gfx1250) — compile-verified
//
#include <hip/hip_runtime.h>
#include <hip/hip_bf16.h>

typedef float v2f __attribute__((ext_vector_type(2)));
typedef float v8f __attribute__((ext_vector_type(8)));

#define NBOX   8400
#define NPAD   8448          // 528 * 16, multiple of 32
#define NSORT  16384         // next pow2 >= NBOX
#define NTILE  528           // NPAD / 16
#define NW16   528           // uint16 words per mask row
#define NW32   264           // uint32 words per mask row

// ---------------------------------------------------------------- sort keys
__global__ void pp_build_keys(const float* __restrict__ x,
                              unsigned long long* __restrict__ keys) {
    int t = blockIdx.x * blockDim.x + threadIdx.x;
    if (t >= NSORT) return;
    unsigned long long key = 0ull;
    if (t < NBOX) {
        float s = x[4 * NBOX + t];
        unsigned int b = __float_as_uint(s);
        // monotonic map: descending float order == descending mapped-uint order
        b = (b & 0x80000000u) ? ~b : (b | 0x80000000u);
        key = ((unsigned long long)b << 32) | (unsigned long long)(0xFFFFFFFFu - (unsigned)t);
    }
    keys[t] = key;   // pads (key==0) sort to the end of a descending sort
}

// one bitonic compare-exchange pass over NSORT keys (final order: descending)
__global__ void pp_bitonic_pass(unsigned long long* __restrict__ keys, int k, int j) {
    int t = blockIdx.x * blockDim.x + threadIdx.x;
    int ixj = t ^ j;
    if (ixj > t && t < NSORT) {
        unsigned long long a = keys[t];
        unsigned long long b = keys[ixj];
        bool up = (t & k) == 0;               // descending segment
        bool sw = up ? (a < b) : (a > b);
        if (sw) { keys[t] = b; keys[ixj] = a; }
    }
}

// ---------------------------------------------------------------- gather
__global__ void pp_gather(const float* __restrict__ x,
                          const unsigned long long* __restrict__ keys,
                          float* __restrict__ b4,      // NPAD*4
                          float* __restrict__ s) {     // NPAD
    int t = blockIdx.x * blockDim.x + threadIdx.x;
    if (t >= NPAD) return;
    if (t < NBOX) {
        unsigned idx = 0xFFFFFFFFu - (unsigned)(keys[t] & 0xFFFFFFFFull);
        b4[t * 4 + 0] = x[0 * NBOX + idx];
        b4[t * 4 + 1] = x[1 * NBOX + idx];
        b4[t * 4 + 2] = x[2 * NBOX + idx];
        b4[t * 4 + 3] = x[3 * NBOX + idx];
        s[t] = x[4 * NBOX + idx];
    } else {
        // zero-area pad box: IoU with anything is 0 -> never suppresses
        b4[t * 4 + 0] = 0.0f; b4[t * 4 + 1] = 0.0f;
        b4[t * 4 + 2] = 0.0f; b4[t * 4 + 3] = 0.0f;
        s[t] = -1.0f;
    }
}

// ------------------------------------------------- pairwise suppression mask
// One wave per 16x16 tile. V_WMMA_F32_16X16X4_F32 computes the union term
// S[i,j] = area_i + area_j as a rank-2 outer product; VALU does intersection.
// C/D layout: VGPR v -> lanes0-15 row M=v, lanes16-31 row M=v+8, N=lane&15.
__global__ void __launch_bounds__(256)
pp_iou_mask(const float4* __restrict__ b4, unsigned short* __restrict__ mask16) {
    int wid  = blockIdx.x * (blockDim.x >> 5) + (threadIdx.x >> 5);
    int lane = threadIdx.x & 31;
    int ti = wid / NTILE;
    int tj = wid - ti * NTILE;
    int l    = lane & 15;
    int half = lane >> 4;

    float4 bi_l = b4[ti * 16 + l];
    float4 bj   = b4[tj * 16 + l];
    float areaI = (bi_l.z - bi_l.x) * (bi_l.w - bi_l.y);
    float areaJ = (bj.z   - bj.x)   * (bj.w   - bj.y);

    // A(16x4): col0 = area_i, col1 = 1 (K=0,1 live in lanes0-15; K=2,3 zero)
    // B(4x16): row0 = 1, row1 = area_j
    v2f A, B;
    A.x = half ? 0.0f : areaI;  A.y = half ? 0.0f : 1.0f;
    B.x = half ? 0.0f : 1.0f;   B.y = half ? 0.0f : areaJ;
    v8f C = {0.f,0.f,0.f,0.f,0.f,0.f,0.f,0.f};
    v8f S = __builtin_amdgcn_wmma_f32_16x16x4_f32(false, A, false, B,
                                                  (short)0, C, false, false);

    int j = tj * 16 + l;
    #pragma unroll
    for (int v = 0; v < 8; ++v) {
        int i = ti * 16 + v + half * 8;
        float4 bi = b4[i];
        float iw = fminf(bi.z, bj.z) - fmaxf(bi.x, bj.x); iw = fmaxf(iw, 0.0f);
        float ih = fminf(bi.w, bj.w) - fmaxf(bi.y, bj.y); ih = fmaxf(ih, 0.0f);
        float inter = iw * ih;
        float iou = inter / (S[v] - inter + 1e-9f);
        int pred = (iou > 0.5f) && (j > i);
        unsigned int w = (unsigned int)__ballot(pred);   // wave32: low 32 bits
        int rlo = ti * 16 + v;
        if (lane == 0) mask16[(size_t)rlo * NW16 + tj]       = (unsigned short)(w & 0xFFFFu);
        if (lane == 1) mask16[(size_t)(rlo + 8) * NW16 + tj] = (unsigned short)(w >> 16);
    }
}

// ------------------------------------------------- sequential greedy reduce
__global__ void __launch_bounds__(288)
pp_nms_reduce(const unsigned short* __restrict__ mask16,
              unsigned char* __restrict__ keep) {
    __shared__ unsigned int remv[NW32];
    int t = threadIdx.x;
    if (t < NW32) remv[t] = 0u;
    __syncthreads();
    for (int i = 0; i < NBOX; ++i) {
        unsigned int r = remv[i >> 5];
        bool removed = (r >> (i & 31)) & 1u;
        __syncthreads();                    // all reads before this round's ORs
        if (!removed) {
            const unsigned int* row =
                (const unsigned int*)(mask16 + (size_t)i * NW16);
            if (t < NW32) remv[t] |= row[t];
        }
        if (t == 0) keep[i] = removed ? 0 : 1;
        __syncthreads();
    }
}

// ---------------------------------------------------------------- output
__global__ void pp_write_out(const float* __restrict__ b4,
                             const float* __restrict__ s,
                             const unsigned char* __restrict__ keep,
                             float* __restrict__ out) {
    int i = blockIdx.x * blockDim.x + threadIdx.x;
    if (i >= NBOX) return;
    float m = (keep[i] && (s[i] > 0.5f)) ? 1.0f : 0.0f;
    out[i * 5 + 0] = b4[i * 4 + 0] * m;
    out[i * 5 + 1] = b4[i * 4 + 1] * m;
    out[i * 5 + 2] = b4[i * 4 + 2] * m;
    out[i * 5 + 3] = b4[i * 4 + 3] * m;
    out[i * 5 + 4] = s[i] * m;
}

extern "C" void kernel_launch(void* const* d_in, const int* in_sizes, int n_in,
                              void* d_out, int out_size, void* d_ws, size_t ws_size,
                              hipStream_t stream) {
    (void)in_sizes; (void)n_in; (void)out_size; (void)ws_size;
    const float* x = (const float*)d_in[0];          // (1, 84, 8400) f32
    float* out = (float*)d_out;                      // 8400*5 f32

    char* ws = (char*)d_ws;
    size_t off = 0;
    unsigned long long* keys = (unsigned long long*)(ws + off); off += (size_t)NSORT * 8;   // 131072
    float* b4 = (float*)(ws + off);                  off += (size_t)NPAD * 4 * 4;           // 135168
    float* s  = (float*)(ws + off);                  off += (size_t)NPAD * 4;               //  33792
    unsigned short* mask16 = (unsigned short*)(ws + off); off += (size_t)NPAD * NW16 * 2;   // ~8.9MB
    unsigned char* keep = (unsigned char*)(ws + off); off += (size_t)NPAD;

    pp_build_keys<<<NSORT / 256, 256, 0, stream>>>(x, keys);

    for (int k = 2; k <= NSORT; k <<= 1)
        for (int j = k >> 1; j > 0; j >>= 1)
            pp_bitonic_pass<<<NSORT / 256, 256, 0, stream>>>(keys, k, j);

    pp_gather<<<NPAD / 256, 256, 0, stream>>>(x, keys, b4, s);

    int tiles = NTILE * NTILE;                 // 278784 waves, 8 waves/block
    pp_iou_mask<<<tiles / 8, 256, 0, stream>>>((const float4*)b4, mask16);

    pp_nms_reduce<<<1, 288, 0, stream>>>(mask16, keep);

    pp_write_out<<<(NBOX + 255) / 256, 256, 0, stream>>>(b4, s, keep, out);
}